// ProviderGAT_56642028699935
// MI455X (gfx1250) — compile-verified
//
#include <hip/hip_runtime.h>
#include <hip/hip_bf16.h>

#define NCODES 262144
#define CDIM   256
#define DDIM   256
#define NH     4
#define HIDDEN 256
#define NSPEC  128

typedef float v2f __attribute__((ext_vector_type(2)));
typedef float v8f __attribute__((ext_vector_type(8)));

// ---- workspace layout (float offsets) ----
#define WS_BBUF    0          // 4096   : B operand of pass1, pre-swizzled per-lane layout
#define WS_VBUF    4096       // 1024   : v[h][c] = W[h,c,:].a2[h]
#define WS_PP      5120       // 1024   : provider[c] * (W[h,c,:].a1[h])
#define WS_SCOREP  6144       // 4
#define WS_GMAX    6148       // 4 (uint keys)
#define WS_SUMEXP  6152       // 4
#define WS_EBUF    8192       // 4*262144 : e[h][n] after leakyrelu
#define WS_PWC     1056768    // 512*1024 : per-block partial weighted_code
#define WS_PSUM    1581056    // 512*4    : per-block partial sum(exp)
#define WS_WC2     1583104    // 4*8*256  : segment-reduced weighted_code
#define WS_PHID    1591296    // 8*256    : partial hidden
// total ~1.59M floats (~6.1 MiB)

__device__ __forceinline__ unsigned fkey(float f) {           // order-preserving float->uint
  unsigned b = __float_as_uint(f);
  return (b & 0x80000000u) ? ~b : (b | 0x80000000u);
}
__device__ __forceinline__ float fdec(unsigned u) {
  unsigned b = (u & 0x80000000u) ? (u ^ 0x80000000u) : ~u;
  return __uint_as_float(b);
}
__device__ __forceinline__ v8f wmma4(v2f a, v2f b, v8f c) {
  return __builtin_amdgcn_wmma_f32_16x16x4_f32(false, a, false, b, (short)0, c, false, false);
}

// ---- setup 1: v[h][c], provider-side partials (1024 blocks x 64 thr) ----
__global__ void kA1(const float* __restrict__ prov, const float* __restrict__ W,
                    const float* __restrict__ a1, const float* __restrict__ a2,
                    float* __restrict__ ws) {
  int h = blockIdx.x >> 8;
  int c = blockIdx.x & 255;
  int t = threadIdx.x;
  const float* wrow = W + (h * CDIM + c) * DDIM;
  const float* A1 = a1 + h * DDIM;
  const float* A2 = a2 + h * DDIM;
  float u = 0.f, v = 0.f;
  for (int d = t; d < DDIM; d += 64) {
    float w = wrow[d];
    u += w * A1[d];
    v += w * A2[d];
  }
  __shared__ float su[64], sv[64];
  su[t] = u; sv[t] = v;
  __syncthreads();
  for (int s = 32; s > 0; s >>= 1) {
    if (t < s) { su[t] += su[t + s]; sv[t] += sv[t + s]; }
    __syncthreads();
  }
  if (t == 0) {
    ws[WS_VBUF + h * CDIM + c] = sv[0];
    ws[WS_PP + h * CDIM + c] = prov[c] * su[0];
  }
}

// ---- setup 2: score_p, gmax init, build swizzled B operand (1 block x 256) ----
__global__ void kA2(float* __restrict__ ws) {
  int t = threadIdx.x;
  if (t < NH) {
    float s = 0.f;
    for (int c = 0; c < CDIM; ++c) s += ws[WS_PP + t * CDIM + c];
    ws[WS_SCOREP + t] = s;
    ((unsigned*)(ws + WS_GMAX))[t] = fkey(-__builtin_inff());
  }
  // Bbuf[kc*64 + l*2 + j] = Vpad[4*kc + 2*(l/16) + j][l%16]
  for (int idx = t * 16; idx < t * 16 + 16; ++idx) {
    int kc = idx >> 6, rem = idx & 63;
    int l = rem >> 1, j = rem & 1;
    int k = 4 * kc + 2 * (l >> 4) + j;
    int hh = l & 15;
    ws[WS_BBUF + idx] = (hh < NH) ? ws[WS_VBUF + hh * CDIM + k] : 0.f;
  }
}

// ---- pass 1: e[h][n] = leakyrelu(score_p + code.v), via WMMA (512 blocks x 256) ----
__global__ __launch_bounds__(256) void kP1(const float* __restrict__ code,
                                           float* __restrict__ ws) {
  __shared__ float ldsB[4096];
  int tid = threadIdx.x;
  for (int i = tid; i < 4096; i += 256) ldsB[i] = ws[WS_BBUF + i];
  __syncthreads();
  int lane = tid & 31, widx = tid >> 5;
  int half = lane >> 4, hm = lane & 15;
  int hs = hm < NH ? hm : 0;
  float sp = ws[WS_SCOREP + hs];
  float* Ebuf = ws + WS_EBUF;
  int gw = blockIdx.x * 8 + widx;
  float lmax = -__builtin_inff();
  for (int i = 0; i < 4; ++i) {
    int n0 = (gw * 4 + i) * 16;
    const float* arow = code + (n0 + hm) * CDIM + 2 * half;  // A: M=hm, K=2*half+{0,1}
    v8f acc = {};
#pragma unroll 8
    for (int kc = 0; kc < 64; ++kc) {
      v2f a = *reinterpret_cast<const v2f*>(arow + kc * 4);
      v2f b = *reinterpret_cast<const v2f*>(&ldsB[kc * 64 + lane * 2]);
      acc = wmma4(a, b, acc);
    }
    if (hm < NH) {  // D: M = j + 8*half (code row), N = hm (head)
      float* ep = Ebuf + hm * NCODES + n0 + 8 * half;
#pragma unroll
      for (int j = 0; j < 8; ++j) {
        float e = acc[j] + sp;
        e = e > 0.f ? e : 0.2f * e;
        ep[j] = e;
        lmax = fmaxf(lmax, e);
      }
    }
  }
  if (hm < NH) atomicMax((unsigned*)(ws + WS_GMAX) + hm, fkey(lmax));
}

// ---- pass 2: weighted_code partials + sumexp partials, via WMMA (512 blocks x 256) ----
__global__ __launch_bounds__(256) void kP2(const float* __restrict__ code,
                                           float* __restrict__ ws) {
  int tid = threadIdx.x;
  int lane = tid & 31, widx = tid >> 5;
  int half = lane >> 4, hm = lane & 15;
  int cs = widx & 3, grp = widx >> 2;  // col slice (4x64), code group (2x256)
  int hs = hm < NH ? hm : 0;
  float gm = fdec(((const unsigned*)(ws + WS_GMAX))[hs]);
  const float* Ebuf = ws + WS_EBUF;
  int base = blockIdx.x * 512 + grp * 256;
  int c0 = cs * 64;
  v8f acc0 = {}, acc1 = {}, acc2 = {}, acc3 = {};
  float sumw = 0.f;
#pragma unroll 4
  for (int t = 0; t < 64; ++t) {
    int n0 = base + t * 4;
    int r = n0 + 2 * half;
    // A: M = hm (head), K = 2*half + {0,1} -> unnormalized softmax weights
    v2f ev = *reinterpret_cast<const v2f*>(Ebuf + hs * NCODES + r);
    v2f a;
    a.x = hm < NH ? __expf(ev.x - gm) : 0.f;
    a.y = hm < NH ? __expf(ev.y - gm) : 0.f;
    sumw += a.x + a.y;
    // B: K = 2*half + {0,1} (code rows), N = hm (column in 16-wide tile)
    const float* p0 = code + r * CDIM + c0 + hm;
    const float* p1 = p0 + CDIM;
    v2f b;
    b.x = p0[0];  b.y = p1[0];  acc0 = wmma4(a, b, acc0);
    b.x = p0[16]; b.y = p1[16]; acc1 = wmma4(a, b, acc1);
    b.x = p0[32]; b.y = p1[32]; acc2 = wmma4(a, b, acc2);
    b.x = p0[48]; b.y = p1[48]; acc3 = wmma4(a, b, acc3);
  }
  __shared__ float wcL[8][NH][64];
  __shared__ float sumL[2][NH][2];
  if (half == 0) {  // D: M = j (head, j<4 valid), N = hm (col)
#pragma unroll
    for (int j = 0; j < NH; ++j) {
      wcL[widx][j][0 + hm]  = acc0[j];
      wcL[widx][j][16 + hm] = acc1[j];
      wcL[widx][j][32 + hm] = acc2[j];
      wcL[widx][j][48 + hm] = acc3[j];
    }
  }
  if (cs == 0 && hm < NH) sumL[grp][hm][half] = sumw;
  __syncthreads();
  int c = tid, csd = c >> 6, cpos = c & 63;
#pragma unroll
  for (int h = 0; h < NH; ++h) {
    float v = wcL[csd][h][cpos] + wcL[4 + csd][h][cpos];
    ws[WS_PWC + blockIdx.x * 1024 + h * 256 + c] = v;
  }
  if (tid < NH) {
    ws[WS_PSUM + blockIdx.x * 4 + tid] =
        sumL[0][tid][0] + sumL[0][tid][1] + sumL[1][tid][0] + sumL[1][tid][1];
  }
}

// ---- reduce block partials (32 blocks x 256) ----
__global__ void kR1(float* __restrict__ ws) {
  int h = blockIdx.x >> 3, seg = blockIdx.x & 7;
  int c = threadIdx.x;
  float s = 0.f;
  int b0 = seg * 64;
  for (int b = 0; b < 64; ++b)
    s += ws[WS_PWC + (b0 + b) * 1024 + h * 256 + c];
  ws[WS_WC2 + (h * 8 + seg) * 256 + c] = s;
  if (blockIdx.x == 0 && c < NH) {
    float t = 0.f;
    for (int b = 0; b < 512; ++b) t += ws[WS_PSUM + b * 4 + c];
    ws[WS_SUMEXP + c] = t;
  }
}

// ---- aggregated = (wc/sumexp) @ W[h] -> combined (32 blocks x 256) ----
__global__ void kR2(const float* __restrict__ W, float* __restrict__ ws,
                    float* __restrict__ out) {
  int h = blockIdx.x >> 3, dseg = blockIdx.x & 7;
  int tid = threadIdx.x;
  __shared__ float wcS[256];
  {
    float s = 0.f;
    for (int seg = 0; seg < 8; ++seg) s += ws[WS_WC2 + (h * 8 + seg) * 256 + tid];
    wcS[tid] = s / ws[WS_SUMEXP + h];
  }
  __syncthreads();
  int dd = tid & 31, part = tid >> 5;
  int d = dseg * 32 + dd;
  float s = 0.f;
  for (int i = 0; i < 32; ++i) {
    int c = part * 32 + i;
    s += wcS[c] * W[(h * CDIM + c) * DDIM + d];
  }
  __shared__ float red[8][32];
  red[part][dd] = s;
  __syncthreads();
  if (part == 0) {
    float tot = 0.f;
#pragma unroll
    for (int p = 0; p < 8; ++p) tot += red[p][dd];
    out[NSPEC + h * DDIM + d] = tot;  // combined
  }
}

// ---- hidden partials: combined @ W1 (8 blocks x 256) ----
__global__ void kR3(const float* __restrict__ W1, const float* __restrict__ out,
                    float* __restrict__ ws) {
  int seg = blockIdx.x;
  int j = threadIdx.x;
  const float* comb = out + NSPEC;
  float s = 0.f;
  int i0 = seg * 128;
  for (int i = 0; i < 128; ++i)
    s += comb[i0 + i] * W1[(i0 + i) * HIDDEN + j];
  ws[WS_PHID + seg * HIDDEN + j] = s;
}

// ---- final: relu + W2 + biases (1 block x 256) ----
__global__ void kR4(const float* __restrict__ b1, const float* __restrict__ W2,
                    const float* __restrict__ b2, const float* __restrict__ ws,
                    float* __restrict__ out) {
  int tid = threadIdx.x;
  __shared__ float hid[HIDDEN];
  float s = b1[tid];
  for (int seg = 0; seg < 8; ++seg) s += ws[WS_PHID + seg * HIDDEN + tid];
  hid[tid] = s > 0.f ? s : 0.f;
  __syncthreads();
  if (tid < NSPEC) {
    float l = b2[tid];
    for (int j = 0; j < HIDDEN; ++j) l += hid[j] * W2[j * NSPEC + tid];
    out[tid] = l;
  }
}

extern "C" void kernel_launch(void* const* d_in, const int* in_sizes, int n_in,
                              void* d_out, int out_size, void* d_ws, size_t ws_size,
                              hipStream_t stream) {
  (void)in_sizes; (void)n_in; (void)out_size; (void)ws_size;
  const float* prov = (const float*)d_in[0];
  const float* code = (const float*)d_in[1];
  const float* W    = (const float*)d_in[2];
  const float* a1   = (const float*)d_in[3];
  const float* a2   = (const float*)d_in[4];
  const float* W1   = (const float*)d_in[5];
  const float* b1   = (const float*)d_in[6];
  const float* W2   = (const float*)d_in[7];
  const float* b2   = (const float*)d_in[8];
  float* out = (float*)d_out;
  float* ws  = (float*)d_ws;

  kA1<<<1024, 64, 0, stream>>>(prov, W, a1, a2, ws);
  kA2<<<1, 256, 0, stream>>>(ws);
  kP1<<<512, 256, 0, stream>>>(code, ws);
  kP2<<<512, 256, 0, stream>>>(code, ws);
  kR1<<<32, 256, 0, stream>>>(ws);
  kR2<<<32, 256, 0, stream>>>(W, ws, out);
  kR3<<<8, 256, 0, stream>>>(W1, out, ws);
  kR4<<<1, 256, 0, stream>>>(b1, W2, b2, ws, out);
}